// GNNModel_19911468384693
// MI455X (gfx1250) — compile-verified
//
#include <hip/hip_runtime.h>

#define N_NODES 100000
#define N_EDGES 1600000
#define D_IN    128
#define D_HID   128
#define D_OUT   64

typedef float v2f __attribute__((ext_vector_type(2)));
typedef float v8f __attribute__((ext_vector_type(8)));

// ---------------------------------------------------------------------------
// Degree / normalization kernels
// ---------------------------------------------------------------------------
__global__ void fill_ones_k(float* __restrict__ p, int n) {
    int i = blockIdx.x * blockDim.x + threadIdx.x;
    if (i < n) p[i] = 1.0f;
}

__global__ void deg_accum_k(const int* __restrict__ dst, float* __restrict__ deg, int e) {
    int i = blockIdx.x * blockDim.x + threadIdx.x;
    if (i < e) atomicAdd(&deg[dst[i]], 1.0f);
}

__global__ void rsqrt_k(float* __restrict__ p, int n) {
    int i = blockIdx.x * blockDim.x + threadIdx.x;
    if (i < n) p[i] = rsqrtf(p[i]);
}

__global__ void edge_norm_k(const int* __restrict__ src, const int* __restrict__ dst,
                            const float* __restrict__ dinv, float* __restrict__ norm, int e) {
    int i = blockIdx.x * blockDim.x + threadIdx.x;
    if (i < e) norm[i] = dinv[src[i]] * dinv[dst[i]];
}

// ---------------------------------------------------------------------------
// FP32 WMMA GEMM: C[M,N] = A[M,K=128] @ B[128,N] (+ optional bias[N])
// Block = 160 threads (5 waves). Each wave owns a 16x64 strip (4 n-tiles).
// grid.x * 5 == M/16 exactly (6250 = 1250 * 5), so no tail guard and EXEC
// stays all-ones for WMMA.
//
// B strip (128 x 64 = 32 KB) is staged in LDS once per block, interleaved in
// (k-pair, col) order so each B fragment is one contiguous ds_load_b64:
//     Bs[(k>>1)*128 + col*2 + (k&1)] = B[k][nBase+col]
// All 32 A fragments (16x4 layout: lanes 0-15 -> K = 4j,4j+1; lanes 16-31 ->
// K = 4j+2,4j+3) are burst-loaded into registers before the MAC loop.
// ---------------------------------------------------------------------------
template <int NT>                    // NT = 4 -> 64-wide strip
__global__ __launch_bounds__(160)
void gemm_wmma_f32_k(const float* __restrict__ A, const float* __restrict__ B,
                     const float* __restrict__ bias, float* __restrict__ C,
                     int N) {
    constexpr int KC = 128;          // K is 128 in every GEMM of this model
    __shared__ float Bs[KC * 64];    // 32 KB

    const int tid   = threadIdx.x;
    const int nBase = blockIdx.y * 64;

    // cooperative, coalesced load of the B strip into LDS (pair-interleaved)
    for (int i = tid; i < KC * 64; i += 160) {
        const int k   = i >> 6;
        const int col = i & 63;
        Bs[(k >> 1) * 128 + col * 2 + (k & 1)] = B[(size_t)k * N + nBase + col];
    }
    __syncthreads();

    const int lane  = tid & 31;
    const int wave  = tid >> 5;
    const int mTile = blockIdx.x * 5 + wave;
    const int half  = lane >> 4;     // 0 or 1
    const int l16   = lane & 15;
    const int row   = mTile * 16 + l16;

    // burst-load all A fragments for this wave's 16-row tile
    const float* Arow = A + (size_t)row * KC + 2 * half;
    v2f afrag[KC / 4];
#pragma unroll
    for (int j = 0; j < KC / 4; ++j)
        afrag[j] = *(const v2f*)(Arow + j * 4);

    v8f acc[NT];
#pragma unroll
    for (int t = 0; t < NT; ++t)
        acc[t] = (v8f){0.f, 0.f, 0.f, 0.f, 0.f, 0.f, 0.f, 0.f};

    // dense WMMA stream: B fragments come from LDS as single b64 loads
#pragma unroll
    for (int j = 0; j < KC / 4; ++j) {
        const float* bp = &Bs[(2 * j + half) * 128 + l16 * 2];
#pragma unroll
        for (int t = 0; t < NT; ++t) {
            v2f b = *(const v2f*)(bp + t * 32);
            acc[t] = __builtin_amdgcn_wmma_f32_16x16x4_f32(
                false, afrag[j], false, b, (short)0, acc[t], false, false);
        }
    }

#pragma unroll
    for (int t = 0; t < NT; ++t) {
        const int col = nBase + t * 16 + l16;
        const float bv = bias ? bias[col] : 0.0f;
#pragma unroll
        for (int r = 0; r < 8; ++r) {
            const int m = mTile * 16 + r + 8 * half;
            C[(size_t)m * N + col] = acc[t][r] + bv;
        }
    }
}

// ---------------------------------------------------------------------------
// agg[i,:] = dinv[i]^2 * h[i,:]   (self-loop message, initializes buffer)
// one thread per float4 (n*32 threads)
// ---------------------------------------------------------------------------
__global__ void self_init_k(const float* __restrict__ h, const float* __restrict__ dinv,
                            float* __restrict__ agg, int n) {
    int i = blockIdx.x * blockDim.x + threadIdx.x;
    if (i >= n * 32) return;
    int node = i >> 5;
    float dv = dinv[node];
    float s  = dv * dv;
    float4 v = ((const float4*)h)[i];
    v.x *= s; v.y *= s; v.z *= s; v.w *= s;
    ((float4*)agg)[i] = v;
}

// ---------------------------------------------------------------------------
// Edge scatter: one wave (32 lanes) per edge; lane handles 4 consecutive feats.
// agg[dst] += norm[e] * h[src]   (h is L2-resident: 51.2 MB < 192 MB L2)
// ---------------------------------------------------------------------------
__global__ void edge_scatter_k(const float* __restrict__ h, const int* __restrict__ src,
                               const int* __restrict__ dst, const float* __restrict__ norm,
                               float* __restrict__ agg, int e) {
    int gid  = blockIdx.x * blockDim.x + threadIdx.x;
    int eidx = gid >> 5;
    int lane = gid & 31;
    if (eidx >= e) return;
    int   s = src[eidx];
    int   d = dst[eidx];
    float w = norm[eidx];
    float4 v = ((const float4*)(h + (size_t)s * D_HID))[lane];
    float* o = agg + (size_t)d * D_HID + lane * 4;
    atomicAdd(o + 0, v.x * w);
    atomicAdd(o + 1, v.y * w);
    atomicAdd(o + 2, v.z * w);
    atomicAdd(o + 3, v.w * w);
}

// ---------------------------------------------------------------------------
// h = relu(h + bias), in place; one thread per float4
// ---------------------------------------------------------------------------
__global__ void bias_relu_k(float* __restrict__ h, const float* __restrict__ b, int n) {
    int i = blockIdx.x * blockDim.x + threadIdx.x;
    if (i >= n * 32) return;
    int fc = (i & 31) * 4;
    float4 v = ((float4*)h)[i];
    v.x = fmaxf(v.x + b[fc + 0], 0.0f);
    v.y = fmaxf(v.y + b[fc + 1], 0.0f);
    v.z = fmaxf(v.z + b[fc + 2], 0.0f);
    v.w = fmaxf(v.w + b[fc + 3], 0.0f);
    ((float4*)h)[i] = v;
}

// ---------------------------------------------------------------------------
// Launch
// ---------------------------------------------------------------------------
static inline size_t align256(size_t x) { return (x + 255) & ~(size_t)255; }

extern "C" void kernel_launch(void* const* d_in, const int* in_sizes, int n_in,
                              void* d_out, int out_size, void* d_ws, size_t ws_size,
                              hipStream_t stream) {
    const float* x   = (const float*)d_in[0];
    const int*   ei  = (const int*)d_in[1];
    const int*   src = ei;
    const int*   dst = ei + N_EDGES;
    const float* W1  = (const float*)d_in[2];
    const float* b1  = (const float*)d_in[3];
    const float* W2  = (const float*)d_in[4];
    const float* b2  = (const float*)d_in[5];
    const float* fcW = (const float*)d_in[6];
    const float* fcb = (const float*)d_in[7];
    float*       out = (float*)d_out;

    // workspace layout
    char* ws = (char*)d_ws;
    size_t off = 0;
    float* dinv = (float*)(ws + off); off += align256((size_t)N_NODES * 4);
    float* norm = (float*)(ws + off); off += align256((size_t)N_EDGES * 4);
    float* bufA = (float*)(ws + off); off += align256((size_t)N_NODES * D_HID * 4);
    float* bufB = (float*)(ws + off); off += align256((size_t)N_NODES * D_HID * 4);
    (void)ws_size; (void)in_sizes; (void)n_in; (void)out_size;

    const int T = 256;
    const int gN    = (N_NODES + T - 1) / T;              // node-wise
    const int gE    = (N_EDGES + T - 1) / T;              // edge-wise
    const int gFeat = (N_NODES * 32 + T - 1) / T;         // float4-wise (n*32)
    const int gScat = (N_EDGES * 32 + T - 1) / T;         // wave-per-edge

    // --- normalization: deg -> dinv -> per-edge norm ---
    fill_ones_k<<<gN, T, 0, stream>>>(dinv, N_NODES);
    deg_accum_k<<<gE, T, 0, stream>>>(dst, dinv, N_EDGES);
    rsqrt_k   <<<gN, T, 0, stream>>>(dinv, N_NODES);
    edge_norm_k<<<gE, T, 0, stream>>>(src, dst, dinv, norm, N_EDGES);

    // --- layer 1: h1 = x @ W1 ; agg ; relu(+b1) ---
    dim3 gemmBlk(160);
    dim3 gemmGrid1(1250, D_HID / 64);   // 6250 m-tiles = 1250 blocks * 5 waves
    gemm_wmma_f32_k<4><<<gemmGrid1, gemmBlk, 0, stream>>>(x, W1, nullptr, bufA, D_HID);
    self_init_k   <<<gFeat, T, 0, stream>>>(bufA, dinv, bufB, N_NODES);
    edge_scatter_k<<<gScat, T, 0, stream>>>(bufA, src, dst, norm, bufB, N_EDGES);
    bias_relu_k   <<<gFeat, T, 0, stream>>>(bufB, b1, N_NODES);

    // --- layer 2: h2 = h @ W2 ; agg ; relu(+b2) ---
    gemm_wmma_f32_k<4><<<gemmGrid1, gemmBlk, 0, stream>>>(bufB, W2, nullptr, bufA, D_HID);
    self_init_k   <<<gFeat, T, 0, stream>>>(bufA, dinv, bufB, N_NODES);
    edge_scatter_k<<<gScat, T, 0, stream>>>(bufA, src, dst, norm, bufB, N_EDGES);
    bias_relu_k   <<<gFeat, T, 0, stream>>>(bufB, b2, N_NODES);

    // --- head: out = h @ fc_W + fc_b ---
    dim3 gemmGrid2(1250, D_OUT / 64);
    gemm_wmma_f32_k<4><<<gemmGrid2, gemmBlk, 0, stream>>>(bufB, fcW, fcb, out, D_OUT);
}